// RegressionGCN_12189117186553
// MI455X (gfx1250) — compile-verified
//
#include <hip/hip_runtime.h>
#include <hip/hip_bf16.h>

typedef float v2f __attribute__((ext_vector_type(2)));
typedef float v8f __attribute__((ext_vector_type(8)));

#define NF_IN 50
#define H1    30
#define H2    25
#define PAD   32   // padded feature pitch (128 bytes -> coalesced 128B rows)

// ---------------------------------------------------------------- degree/norm
__global__ void k_deg_init(float* __restrict__ deg, int n) {
  int i = blockIdx.x * blockDim.x + threadIdx.x;
  if (i < n) deg[i] = 1.0f;               // self-loop
}

__global__ void k_deg_edges(const int* __restrict__ dst, float* __restrict__ deg, int e) {
  int i = blockIdx.x * blockDim.x + threadIdx.x;
  if (i < e) atomicAdd(&deg[dst[i]], 1.0f);
}

__global__ void k_rsqrt_inplace(float* __restrict__ deg, int n) {
  int i = blockIdx.x * blockDim.x + threadIdx.x;
  if (i < n) deg[i] = rsqrtf(deg[i]);     // deg >= 1, safe
}

// ---------------------------------------------------------------- WMMA GEMM
// Out[N x PAD] = A[N x lda](:, 0:KIN) @ W[KIN x NCOL]; pad cols of Out get 0.
// One wave = 16-row x 32-col tile (two f32 16x16x4 accumulator chains).
// B is staged in LDS, TRANSPOSED and ZERO-PADDED: WsT[col][k]. Zero B rows
// beyond KIN annihilate invalid K-terms, so A loads need only address
// clamping (no value masks, no divergent branches).
template<int KIN, int NCOL, bool APADDED>
__global__ void __launch_bounds__(128) k_gemm_wmma(
    const float* __restrict__ A, int lda,
    const float* __restrict__ W,
    float* __restrict__ Out, int n) {
  constexpr int KPAD = (KIN + 3) & ~3;
  __shared__ __align__(16) float WsT[PAD][KPAD];   // [col][k], zero-padded

  for (int i = threadIdx.x; i < PAD * KPAD; i += blockDim.x) {
    const int c  = i / KPAD;
    const int kk = i - c * KPAD;
    WsT[c][kk] = (kk < KIN && c < NCOL) ? W[kk * NCOL + c] : 0.0f;
  }
  __syncthreads();

  const int lane = threadIdx.x & 31;
  const int wave = threadIdx.x >> 5;
  const int tile = blockIdx.x * (blockDim.x >> 5) + wave;
  const int row0 = tile * 16;
  if (row0 >= n) return;                  // wave-uniform (after barrier)

  const int m  = lane & 15;               // M (A) / N (B,C,D) index
  const int hi = lane >> 4;               // K half-select

  const int r  = row0 + m;
  const int rc = (r < n) ? r : (n - 1);   // clamp: OOB rows guarded at store
  const float* __restrict__ Arow = A + (size_t)rc * (size_t)lda;

  v8f acc0 = {};                          // cols 0..15
  v8f acc1 = {};                          // cols 16..31

#pragma unroll
  for (int k = 0; k < KPAD; k += 4) {
    const int ka = k + 2 * hi;
    // Clamp A address at the K tail; zero B rows kill those terms anyway.
    const int kc = APADDED ? ka : ((ka <= KIN - 2) ? ka : (KIN - 2));
    const v2f a  = *(const v2f*)(Arow + kc);          // global_load_b64
    const v2f b0 = *(const v2f*)&WsT[m][ka];          // ds_load_b64
    const v2f b1 = *(const v2f*)&WsT[16 + m][ka];     // ds_load_b64
    acc0 = __builtin_amdgcn_wmma_f32_16x16x4_f32(false, a, false, b0, (short)0, acc0, false, false);
    acc1 = __builtin_amdgcn_wmma_f32_16x16x4_f32(false, a, false, b1, (short)0, acc1, false, false);
  }

  // D layout: VGPR rr -> (M = rr + 8*hi, N = lane&15 [+16 for acc1])
#pragma unroll
  for (int rr = 0; rr < 8; ++rr) {
    const int row = row0 + rr + 8 * hi;
    if (row < n) {
      Out[(size_t)row * PAD + m]      = acc0[rr];
      Out[(size_t)row * PAD + 16 + m] = acc1[rr];
    }
  }
}

// -------------------------------------------------- self-loop init of agg
// agg[i][:] = dinv[i]^2 * H[i][:]   (float4-vectorized stream over N*PAD)
__global__ void k_self_init(const float4* __restrict__ H4, const float* __restrict__ dinv,
                            float4* __restrict__ agg4, int n) {
  int i = blockIdx.x * blockDim.x + threadIdx.x;   // over n * (PAD/4)
  if (i < n * (PAD / 4)) {
    float w = dinv[i >> 3];
    w *= w;
    float4 v = H4[i];
    v.x *= w; v.y *= w; v.z *= w; v.w *= w;
    agg4[i] = v;
  }
}

// -------------------------------------------------- edge gather/scatter
// 8 lanes per edge: each lane owns one float4 of the 128B feature row, so the
// gather is one coalesced 128B transaction per edge; indices broadcast.
template<int NF>
__global__ void k_edge_scatter(const int* __restrict__ src, const int* __restrict__ dst,
                               const float* __restrict__ dinv,
                               const float* __restrict__ H, float* __restrict__ agg, int e) {
  const int t  = blockIdx.x * blockDim.x + threadIdx.x;
  const int ei = t >> 3;
  const int f  = (t & 7) << 2;            // 0,4,...,28
  if (ei >= e) return;
  const int s = src[ei];
  const int d = dst[ei];
  const float w = dinv[s] * dinv[d];
  const float4 v = *(const float4*)(H + (size_t)s * PAD + f);
  float* __restrict__ ad = agg + (size_t)d * PAD + f;
  if (f + 3 < NF) {                       // full quad
    atomicAdd(ad + 0, w * v.x);
    atomicAdd(ad + 1, w * v.y);
    atomicAdd(ad + 2, w * v.z);
    atomicAdd(ad + 3, w * v.w);
  } else if (f < NF) {                    // tail quad (NF=30: f=28 -> 2; NF=25: f=24 -> 1)
    atomicAdd(ad + 0, w * v.x);
    if (f + 1 < NF) atomicAdd(ad + 1, w * v.y);
    if (f + 2 < NF) atomicAdd(ad + 2, w * v.z);
  }
}

// -------------------------------------------------- epilogues
__global__ void k_bias_relu_pad(const float* __restrict__ agg, const float* __restrict__ b,
                                float* __restrict__ out, int n, int nf) {
  int i = blockIdx.x * blockDim.x + threadIdx.x;
  if (i < n * PAD) {
    const int f = i & (PAD - 1);
    const float v = (f < nf) ? (agg[i] + b[f]) : 0.0f;
    out[i] = fmaxf(v, 0.0f);
  }
}

__global__ void k_bias_compact(const float* __restrict__ agg, const float* __restrict__ b,
                               float* __restrict__ out, int n, int nf) {
  int i = blockIdx.x * blockDim.x + threadIdx.x;
  if (i < n * nf) {
    const int node = i / nf;
    const int f    = i - node * nf;
    out[i] = agg[(size_t)node * PAD + f] + b[f];
  }
}

// ---------------------------------------------------------------- launch
extern "C" void kernel_launch(void* const* d_in, const int* in_sizes, int n_in,
                              void* d_out, int out_size, void* d_ws, size_t ws_size,
                              hipStream_t stream) {
  (void)n_in; (void)out_size; (void)ws_size;
  const float* x  = (const float*)d_in[0];
  const int*   ei = (const int*)  d_in[1];   // [2, E]: row0 = src, row1 = dst
  const float* W1 = (const float*)d_in[2];
  const float* b1 = (const float*)d_in[3];
  const float* W2 = (const float*)d_in[4];
  const float* b2 = (const float*)d_in[5];
  float* out = (float*)d_out;

  const int N = in_sizes[0] / NF_IN;
  const int E = in_sizes[1] / 2;
  const int* src = ei;
  const int* dst = ei + E;

  // workspace layout (floats): [dinv: N] [A: N*PAD] [B: N*PAD] [C: N*PAD]
  float* ws   = (float*)d_ws;
  float* dinv = ws;
  float* A    = ws + 131072;                   // keeps A 128B-aligned
  float* B    = A + (size_t)N * PAD;
  float* C    = B + (size_t)N * PAD;

  const int T   = 256;
  const int gN  = (N + T - 1) / T;
  const int gE  = (E + T - 1) / T;
  const int gNQ = (N * (PAD / 4) + T - 1) / T;
  const int gNP = (N * PAD + T - 1) / T;
  const int gE8 = (E * 8 + T - 1) / T;         // 8 lanes per edge
  const int tiles = (N + 15) / 16;
  const int gGemm = (tiles + 3) / 4;           // 4 waves (128 thr) per block

  // norm = D^{-1/2} with self-loops
  k_deg_init     <<<gN, T, 0, stream>>>(dinv, N);
  k_deg_edges    <<<gE, T, 0, stream>>>(dst, dinv, E);
  k_rsqrt_inplace<<<gN, T, 0, stream>>>(dinv, N);

  // ---- layer 1: h1 = relu( scatter(norm * (x@W1)[src]) + b1 )
  k_gemm_wmma<NF_IN, H1, false><<<gGemm, 128, 0, stream>>>(x, NF_IN, W1, A, N);
  k_self_init       <<<gNQ, T, 0, stream>>>((const float4*)A, dinv, (float4*)B, N);
  k_edge_scatter<H1><<<gE8, T, 0, stream>>>(src, dst, dinv, A, B, E);
  k_bias_relu_pad   <<<gNP, T, 0, stream>>>(B, b1, C, N, H1);

  // ---- layer 2: out = scatter(norm * (h1@W2)[src]) + b2  (C is zero-padded)
  k_gemm_wmma<H1, H2, true><<<gGemm, 128, 0, stream>>>(C, PAD, W2, A, N);
  k_self_init       <<<gNQ, T, 0, stream>>>((const float4*)A, dinv, (float4*)B, N);
  k_edge_scatter<H2><<<gE8, T, 0, stream>>>(src, dst, dinv, A, B, E);
  k_bias_compact    <<<(N * H2 + T - 1) / T, T, 0, stream>>>(B, b2, out, N, H2);
}